// FSQ_AdaptiveQuant_1271310319897
// MI455X (gfx1250) — compile-verified
//
#include <hip/hip_runtime.h>
#include <math.h>

// ---------------------------------------------------------------------------
// FSQ adaptive quantization for MI455X (gfx1250, wave32).
// Bandwidth-bound problem (~400 MiB HBM traffic -> ~17us floor @ 23.3 TB/s).
// Both GEMMs run on the matrix pipe via V_WMMA_F32_16X16X4_F32 (exact fp32).
// Round 2: zero-padded Wc LDS tile (no per-step mask mul), bias folded into
// the WMMA C accumulator, deeper K-loop unroll for more loads in flight.
// ---------------------------------------------------------------------------

typedef __attribute__((ext_vector_type(2))) float v2f;
typedef __attribute__((ext_vector_type(8))) float v8f;

#define N_TOK   (64 * 1024)   // B*S
#define C_IN    768
#define D_DIM   8
#define L_LEV   15
#define INV_BW  2.0f          // 1 / bandwidth(0.5)
#define WC_PITCH 772          // 768 + 4 floats: de-conflict LDS banks

// Workspace layout (floats):
//   [0, N_TOK*8)                 zc
//   [N_TOK*8, N_TOK*8+128)       kde_acc  (8 dims x 16 padded levels)
//   [.. +128, .. +256)           scaled   (8 dims x 16 padded levels)

// ===========================================================================
// Kernel 1: compress GEMM (z[N,768] x Wc^T[768,8]) via WMMA f32 16x16x4,
// fused with per-block KDE partial sums.
//
// WMMA fp32 layout (ISA 7.12.2):
//   A 16x4 : lane m (0-15) holds [M=m, K={0,1}] in v0,v1; lane 16+m holds K={2,3}
//   B 4x16 : lane n (0-15) holds [K={0,1}, N=n];          lane 16+n holds K={2,3}
//   C/D    : vgpr r, lane n -> [M=r, N=n]; lane 16+n -> [M=8+r, N=n]
// After the GEMM each lane holds ONE output dim (N=lane%16) for 8 tokens —
// exactly the shape needed for the per-dim KDE reduction.
// ===========================================================================
__launch_bounds__(256)
__global__ void k_compress_kde(const float* __restrict__ z,
                               const float* __restrict__ Wc,
                               const float* __restrict__ bc,
                               const float* __restrict__ centers,
                               float* __restrict__ zc,
                               float* __restrict__ kde_acc)
{
    // 16 rows: rows 8..15 are zero so padded B columns need no runtime mask.
    __shared__ float wc_lds[16 * WC_PITCH];
    __shared__ float kde_lds[D_DIM * 16];

    const int tid  = threadIdx.x;
    const int lane = tid & 31;
    const int wave = tid >> 5;

    if (tid < D_DIM * 16) kde_lds[tid] = 0.0f;
    // Stage Wc [8 x 768] -> LDS rows 0..7; zero rows 8..15. Coalesced.
    for (int i = tid; i < 16 * C_IN; i += 256) {
        int r = i / C_IN, c = i - r * C_IN;
        wc_lds[r * WC_PITCH + c] = (r < D_DIM) ? Wc[i] : 0.0f;
    }
    __syncthreads();

    const int m0     = blockIdx.x * 128 + wave * 16; // first token of this wave
    const int lane16 = lane & 15;
    const int hi     = lane >> 4;                    // 0: K={0,1}, 1: K={2,3}
    const float* zrow  = z + (long)(m0 + lane16) * C_IN + hi * 2;
    const float* wcrow = &wc_lds[lane16 * WC_PITCH + hi * 2];

    const int  dim    = lane16;
    const bool active = dim < D_DIM;
    const float bias  = active ? bc[dim] : 0.0f;

    // Fold the compress bias into the WMMA C accumulator (D = A*B + C):
    // bias is per output column == per lane in the C/D layout.
    v8f acc;
#pragma unroll
    for (int r = 0; r < 8; ++r) acc[r] = bias;

#pragma unroll 8
    for (int k = 0; k < C_IN; k += 4) {
        v2f a = *(const v2f*)(zrow + k);             // global, 8B aligned
        v2f b = *(const v2f*)(wcrow + k);            // LDS,    8B aligned
        acc = __builtin_amdgcn_wmma_f32_16x16x4_f32(
            /*neg_a=*/false, a, /*neg_b=*/false, b,
            /*c_mod=*/(short)0, acc, /*reuse_a=*/false, /*reuse_b=*/false);
    }

    if (active) {
        // Spill zc: lane holds dim `dim` for tokens m0 + r + hi*8.
#pragma unroll
        for (int r = 0; r < 8; ++r)
            zc[(long)(m0 + r + hi * 8) * D_DIM + dim] = acc[r];

        // KDE partials: sum over this lane's 8 tokens for each of 15 centers.
#pragma unroll 1
        for (int l = 0; l < L_LEV; ++l) {
            const float cen = centers[dim * L_LEV + l];
            float s = 0.0f;
#pragma unroll
            for (int r = 0; r < 8; ++r) {
                float d = (cen - acc[r]) * INV_BW;
                s += __expf(-0.5f * d * d);
            }
            atomicAdd(&kde_lds[dim * 16 + l], s);
        }
    }
    __syncthreads();
    if (tid < D_DIM * 16) {
        float v = kde_lds[tid];
        if (v != 0.0f) atomicAdd(&kde_acc[tid], v);
    }
}

// ===========================================================================
// Kernel 2: normalize KDE -> adaptive scaled centers [8 x 16(pad)].
// Also writes the scalar second output (0.0f).
// ===========================================================================
__global__ void k_scale(const float* __restrict__ kde_acc,
                        const float* __restrict__ centers,
                        float* __restrict__ scaled,
                        float* __restrict__ aux_out)
{
    __shared__ float w_lds[D_DIM * 16];
    __shared__ float rowsum[D_DIM];

    const int t = threadIdx.x;            // 128 threads
    const int d = t >> 4, l = t & 15;

    float w = 0.0f;
    if (l < L_LEV)
        w = kde_acc[t] * (1.0f / (float)N_TOK) + 1e-6f;
    w_lds[t] = w;
    __syncthreads();

    if (t < D_DIM) {
        float s = 0.0f;
        for (int i = 0; i < L_LEV; ++i) s += w_lds[t * 16 + i];
        rowsum[t] = s;
    }
    __syncthreads();

    scaled[t] = (l < L_LEV) ? centers[d * L_LEV + l] * (w / rowsum[d]) : 0.0f;
    if (t == 0) aux_out[0] = 0.0f;
}

// ===========================================================================
// Kernel 3: quantize (argmin over 15 scaled centers, first-min tie-break to
// match jnp.argmin) + expand GEMM (q[N,8] x We^T[8,768]) via 2 chained
// WMMA f32 16x16x4 steps per 16x16 output tile, bias pre-loaded into C.
// ===========================================================================
__launch_bounds__(256)
__global__ void k_quant_expand(const float* __restrict__ zc,
                               const float* __restrict__ scaled,
                               const float* __restrict__ We,  // [768][8] row-major
                               const float* __restrict__ be,
                               float* __restrict__ out)
{
    __shared__ float q_lds[128 * D_DIM];   // 128 tokens x 8 dims
    __shared__ float s_lds[D_DIM * 16];
    __shared__ float be_lds[C_IN];

    const int tid  = threadIdx.x;
    const int lane = tid & 31;
    const int wave = tid >> 5;
    const long tok0 = (long)blockIdx.x * 128;

    if (tid < D_DIM * 16) s_lds[tid] = scaled[tid];
    for (int i = tid; i < C_IN; i += 256) be_lds[i] = be[i];
    __syncthreads();

    // ---- Quantize 128 tokens x 8 dims (coalesced zc reads) ----
#pragma unroll
    for (int i = 0; i < 4; ++i) {
        const int idx = tid + i * 256;         // 0..1023
        const int dim = idx & 7;
        const float zv = zc[tok0 * D_DIM + idx];
        float best = s_lds[dim * 16];
        float bd   = fabsf(zv - best);
#pragma unroll 1
        for (int l = 1; l < L_LEV; ++l) {
            float c  = s_lds[dim * 16 + l];
            float dd = fabsf(zv - c);
            if (dd < bd) { bd = dd; best = c; } // strict '<' => first min wins
        }
        q_lds[idx] = best;
    }
    __syncthreads();

    // ---- Expand: each wave owns 16 tokens ----
    const int m0     = wave * 16;
    const int lane16 = lane & 15;
    const int hi     = lane >> 4;
    const float* qbase = &q_lds[m0 * D_DIM];

    // A fragments (from LDS): kstep0 -> K = hi*2 + {0,1}, kstep1 -> +4.
    const v2f a0 = *(const v2f*)(qbase + lane16 * D_DIM + hi * 2);
    const v2f a1 = *(const v2f*)(qbase + lane16 * D_DIM + 4 + hi * 2);

    float* orow = out + (tok0 + m0) * (long)C_IN;

#pragma unroll 2
    for (int j = 0; j < C_IN / 16; ++j) {
        const int c = j * 16 + lane16;                 // output column
        const float* wrow = We + c * D_DIM + hi * 2;   // B[k][n] = We[c][k]
        const v2f b0 = *(const v2f*)(wrow);
        const v2f b1 = *(const v2f*)(wrow + 4);

        // Fold expand bias (per column == per lane) into the C accumulator.
        const float bias = be_lds[c];
        v8f acc;
#pragma unroll
        for (int r = 0; r < 8; ++r) acc[r] = bias;

        acc = __builtin_amdgcn_wmma_f32_16x16x4_f32(false, a0, false, b0,
                                                    (short)0, acc, false, false);
        acc = __builtin_amdgcn_wmma_f32_16x16x4_f32(false, a1, false, b1,
                                                    (short)0, acc, false, false);

#pragma unroll
        for (int r = 0; r < 8; ++r)
            orow[(long)(r + hi * 8) * C_IN + c] = acc[r];
    }
}

// ===========================================================================
extern "C" void kernel_launch(void* const* d_in, const int* in_sizes, int n_in,
                              void* d_out, int out_size, void* d_ws, size_t ws_size,
                              hipStream_t stream)
{
    const float* z       = (const float*)d_in[0];   // [N_TOK, 768]
    const float* Wc      = (const float*)d_in[1];   // [8, 768]
    const float* bc      = (const float*)d_in[2];   // [8]
    const float* We      = (const float*)d_in[3];   // [768, 8]
    const float* be      = (const float*)d_in[4];   // [768]
    const float* centers = (const float*)d_in[5];   // [8, 15]

    float* out = (float*)d_out;                     // [N_TOK*768] + 1 aux scalar

    float* zc      = (float*)d_ws;                  // N_TOK*8 floats
    float* kde_acc = zc + (size_t)N_TOK * D_DIM;    // 128 floats
    float* scaled  = kde_acc + 128;                 // 128 floats

    hipMemsetAsync(kde_acc, 0, 128 * sizeof(float), stream);

    k_compress_kde<<<N_TOK / 128, 256, 0, stream>>>(z, Wc, bc, centers, zc, kde_acc);
    k_scale<<<1, 128, 0, stream>>>(kde_acc, centers, scaled,
                                   out + (size_t)N_TOK * C_IN);
    k_quant_expand<<<N_TOK / 128, 256, 0, stream>>>(zc, scaled, We, be, out);
}